// PeptideMHCPredictor_57449482551355
// MI455X (gfx1250) — compile-verified
//
#include <hip/hip_runtime.h>
#include <hip/hip_bf16.h>
#include <math.h>

#define EPS 1e-5f

typedef __attribute__((ext_vector_type(2))) float v2f;
typedef __attribute__((ext_vector_type(8))) float v8f;

// ---------------------------------------------------------------- utilities
__global__ void zero_f32(float* __restrict__ p, long long n) {
    long long i = (long long)blockIdx.x * blockDim.x + threadIdx.x;
    if (i < n) p[i] = 0.0f;
}

// deg count: cnt[dst[e]] += 1
__global__ void deg_count(const int* __restrict__ dst, float* __restrict__ cnt, int E) {
    int e = blockIdx.x * blockDim.x + threadIdx.x;
    if (e < E) atomicAdd(&cnt[dst[e]], 1.0f);
}

// dinv[i] = rsqrt(cnt[i] + 1)  (in place)
__global__ void deg_finalize(float* __restrict__ d, int N) {
    int i = blockIdx.x * blockDim.x + threadIdx.x;
    if (i < N) d[i] = rsqrtf(d[i] + 1.0f);
}

// h[i][f] = sum_{k<3} x[i*3+k] * W1[k*64+f]
__global__ void lin1(const float* __restrict__ x, const float* __restrict__ W1,
                     float* __restrict__ h, int N) {
    int idx = blockIdx.x * blockDim.x + threadIdx.x;
    if (idx >= N * 64) return;
    int i = idx >> 6, f = idx & 63;
    const float* xr = x + i * 3;
    h[idx] = xr[0] * W1[f] + xr[1] * W1[64 + f] + xr[2] * W1[128 + f];
}

// agg[dst][f] += h[src][f] * dinv[src] * dinv[dst]
__global__ void edge_agg(const float* __restrict__ h, const int* __restrict__ src,
                         const int* __restrict__ dst, const float* __restrict__ dinv,
                         float* __restrict__ agg, long long E) {
    long long idx = (long long)blockIdx.x * blockDim.x + threadIdx.x;
    if (idx >= E * 64) return;
    int e = (int)(idx >> 6), f = (int)(idx & 63);
    int s = src[e], d = dst[e];
    float c = dinv[s] * dinv[d];
    atomicAdd(&agg[(long long)d * 64 + f], h[(long long)s * 64 + f] * c);
}

// agg[i][f] += h[i][f] * dinv[i]^2 + b[f]
__global__ void self_bias(float* __restrict__ agg, const float* __restrict__ h,
                          const float* __restrict__ dinv, const float* __restrict__ b, int N) {
    int idx = blockIdx.x * blockDim.x + threadIdx.x;
    if (idx >= N * 64) return;
    int i = idx >> 6, f = idx & 63;
    float di = dinv[i];
    agg[idx] += h[idx] * di * di + b[f];
}

// per-feature sum / sumsq over nodes: stats[0..63]=sum, stats[64..127]=sumsq
__global__ void bn_stats(const float* __restrict__ X, float* __restrict__ stats, int N) {
    __shared__ float sh[512];
    int tid = threadIdx.x;                 // 256 threads
    int f = tid & 63, r = tid >> 6;        // 4 row-lanes per feature
    float s = 0.0f, s2 = 0.0f;
    for (int i = blockIdx.x * 4 + r; i < N; i += gridDim.x * 4) {
        float v = X[(long long)i * 64 + f];
        s += v; s2 += v * v;
    }
    sh[tid] = s; sh[256 + tid] = s2;
    __syncthreads();
    if (tid < 64) {
        float ts  = sh[tid] + sh[tid + 64] + sh[tid + 128] + sh[tid + 192];
        float ts2 = sh[256 + tid] + sh[320 + tid] + sh[384 + tid] + sh[448 + tid];
        atomicAdd(&stats[tid], ts);
        atomicAdd(&stats[64 + tid], ts2);
    }
}

// out[i][f] = relu(gamma*(x-mu)*rsqrt(var+eps)+beta)
__global__ void bn_apply_relu(const float* __restrict__ X, const float* __restrict__ stats,
                              const float* __restrict__ gamma, const float* __restrict__ beta,
                              float* __restrict__ out, int N) {
    int idx = blockIdx.x * blockDim.x + threadIdx.x;
    if (idx >= N * 64) return;
    int f = idx & 63;
    float inv_n = 1.0f / (float)N;
    float mu = stats[f] * inv_n;
    float var = stats[64 + f] * inv_n - mu * mu;
    float y = gamma[f] * (X[idx] - mu) * rsqrtf(var + EPS) + beta[f];
    out[idx] = fmaxf(y, 0.0f);
}

// OUT[M,64] = IN[M,64] @ W[64,64] (+bias)(+relu), via V_WMMA_F32_16X16X4_F32.
// One wave computes one 16x16 tile; K=64 -> 16 chained WMMAs.
__global__ void gemm64_wmma(const float* __restrict__ IN, const float* __restrict__ W,
                            const float* __restrict__ bias, float* __restrict__ OUT,
                            int M, int relu) {
    int lane = threadIdx.x & 31;
    int wave = threadIdx.x >> 5;
    int tile = blockIdx.x * 4 + wave;            // 4 waves per block
    int mTiles = (M + 15) >> 4;
    if (tile >= mTiles * 4) return;              // wave-uniform exit: EXEC stays all-1s
    int m0 = (tile >> 2) * 16;
    int n0 = (tile & 3) * 16;
    int r  = lane & 15;
    int kh = lane >> 4;                          // 0: K pair {k,k+1}; 1: {k+2,k+3}

    int arow = m0 + r;
    if (arow >= M) arow = M - 1;                 // clamp loads (never hit: M%16==0)

    v8f acc = {0.f, 0.f, 0.f, 0.f, 0.f, 0.f, 0.f, 0.f};
    #pragma unroll
    for (int k = 0; k < 64; k += 4) {
        int ka = k + kh * 2;
        v2f a, b;
        a.x = IN[(long long)arow * 64 + ka];
        a.y = IN[(long long)arow * 64 + ka + 1];
        b.x = W[ka * 64 + n0 + r];
        b.y = W[(ka + 1) * 64 + n0 + r];
        acc = __builtin_amdgcn_wmma_f32_16x16x4_f32(false, a, false, b,
                                                    (short)0, acc, false, false);
    }
    float bv = bias ? bias[n0 + r] : 0.0f;
    #pragma unroll
    for (int v = 0; v < 8; ++v) {
        int row = m0 + kh * 8 + v;               // C/D layout: VGPR v -> M = v + 8*(lane>=16)
        if (row < M) {
            float val = acc[v] + bv;
            if (relu) val = fmaxf(val, 0.0f);
            OUT[(long long)row * 64 + n0 + r] = val;
        }
    }
}

// poolSum[g][f] += h[i][f]; poolCnt[g] += 1 (once per node)
__global__ void pool_sum(const float* __restrict__ h, const int* __restrict__ batch,
                         float* __restrict__ poolSum, float* __restrict__ poolCnt, int N) {
    int idx = blockIdx.x * blockDim.x + threadIdx.x;
    if (idx >= N * 64) return;
    int i = idx >> 6, f = idx & 63;
    int g = batch[i];
    atomicAdd(&poolSum[(long long)g * 64 + f], h[idx]);
    if (f == 0) atomicAdd(&poolCnt[g], 1.0f);
}

__global__ void pool_div(const float* __restrict__ poolSum, const float* __restrict__ poolCnt,
                         float* __restrict__ pooled, int G) {
    int idx = blockIdx.x * blockDim.x + threadIdx.x;
    if (idx >= G * 64) return;
    int g = idx >> 6;
    pooled[idx] = poolSum[idx] / fmaxf(poolCnt[g], 1.0f);
}

// out[g] = sigmoid( dot(z[g,:], fcW2[:,0]) + fcb2 )
__global__ void head(const float* __restrict__ z, const float* __restrict__ fcW2,
                     const float* __restrict__ fcb2, float* __restrict__ out, int G) {
    int g = blockIdx.x * blockDim.x + threadIdx.x;
    if (g >= G) return;
    float acc = fcb2[0];
    const float* zr = z + (long long)g * 64;
    #pragma unroll 8
    for (int f = 0; f < 64; ++f) acc += zr[f] * fcW2[f];
    out[g] = 1.0f / (1.0f + expf(-acc));
}

// ---------------------------------------------------------------- launch
static inline int cdiv(long long a, long long b) { return (int)((a + b - 1) / b); }

extern "C" void kernel_launch(void* const* d_in, const int* in_sizes, int n_in,
                              void* d_out, int out_size, void* d_ws, size_t ws_size,
                              hipStream_t stream) {
    const float* x      = (const float*)d_in[0];
    const int*   src    = (const int*)  d_in[1];
    const int*   dst    = (const int*)  d_in[2];
    const int*   batch  = (const int*)  d_in[3];
    const float* W1     = (const float*)d_in[5];
    const float* b1     = (const float*)d_in[6];
    const float* gamma1 = (const float*)d_in[7];
    const float* beta1  = (const float*)d_in[8];
    const float* W2     = (const float*)d_in[9];
    const float* b2     = (const float*)d_in[10];
    const float* gamma2 = (const float*)d_in[11];
    const float* beta2  = (const float*)d_in[12];
    const float* fcW1   = (const float*)d_in[13];
    const float* fcb1   = (const float*)d_in[14];
    const float* fcW2   = (const float*)d_in[15];
    const float* fcb2   = (const float*)d_in[16];
    float* out = (float*)d_out;

    const int N = in_sizes[0] / 3;
    const long long E = in_sizes[1];
    const int G = out_size;              // output is [G,1]
    const long long NF = (long long)N * 64;

    // workspace layout (floats)
    float* ws = (float*)d_ws;
    long long off = 0;
    float* dinv    = ws + off; off += N;
    float* A       = ws + off; off += NF;        // h / h1n / agg2
    float* B       = ws + off; off += NF;        // agg1 / h2 / h2n
    float* stats   = ws + off; off += 128;
    float* poolSum = ws + off; off += (long long)G * 64;
    float* poolCnt = ws + off; off += G;
    float* pooled  = ws + off; off += (long long)G * 64;
    float* zbuf    = ws + off; off += (long long)G * 64;
    const long long totalF = off;

    const int T = 256;
    // zero everything we use (deg counts, agg buffers, stats, pools)
    zero_f32<<<cdiv(totalF, T), T, 0, stream>>>(ws, totalF);

    // degree + dinv
    deg_count<<<cdiv(E, T), T, 0, stream>>>(dst, dinv, (int)E);
    deg_finalize<<<cdiv(N, T), T, 0, stream>>>(dinv, N);

    // ---- layer 1 ----
    lin1<<<cdiv(NF, T), T, 0, stream>>>(x, W1, A, N);
    edge_agg<<<cdiv(E * 64, T), T, 0, stream>>>(A, src, dst, dinv, B, E);
    self_bias<<<cdiv(NF, T), T, 0, stream>>>(B, A, dinv, b1, N);
    bn_stats<<<512, T, 0, stream>>>(B, stats, N);
    bn_apply_relu<<<cdiv(NF, T), T, 0, stream>>>(B, stats, gamma1, beta1, A, N);

    // ---- layer 2 ----
    int mTilesN = (N + 15) >> 4;
    gemm64_wmma<<<cdiv((long long)mTilesN * 4, 4), 128, 0, stream>>>(A, W2, nullptr, B, N, 0);
    zero_f32<<<cdiv(NF, T), T, 0, stream>>>(A, NF);
    zero_f32<<<1, 128, 0, stream>>>(stats, 128);
    edge_agg<<<cdiv(E * 64, T), T, 0, stream>>>(B, src, dst, dinv, A, E);
    self_bias<<<cdiv(NF, T), T, 0, stream>>>(A, B, dinv, b2, N);
    bn_stats<<<512, T, 0, stream>>>(A, stats, N);
    bn_apply_relu<<<cdiv(NF, T), T, 0, stream>>>(A, stats, gamma2, beta2, B, N);

    // ---- pooling ----
    pool_sum<<<cdiv(NF, T), T, 0, stream>>>(B, batch, poolSum, poolCnt, N);
    pool_div<<<cdiv((long long)G * 64, T), T, 0, stream>>>(poolSum, poolCnt, pooled, G);

    // ---- MLP head ----
    int mTilesG = (G + 15) >> 4;
    gemm64_wmma<<<cdiv((long long)mTilesG * 4, 4), 128, 0, stream>>>(pooled, fcW1, fcb1, zbuf, G, 1);
    head<<<cdiv(G, T), T, 0, stream>>>(zbuf, fcW2, fcb2, out, G);
}